// ModulatedConv2d_80693845557881
// MI455X (gfx1250) — compile-verified
//
#include <hip/hip_runtime.h>
#include <stdint.h>

typedef float v2f __attribute__((ext_vector_type(2)));
typedef float v8f __attribute__((ext_vector_type(8)));

#define B_    16
#define CIN   512
#define COUT  512
#define KDIM  (CIN * 9)      // 4608
#define NN    4096           // 64*64

#define BM 128
#define BN 128
#define BK 32
#define KTILES (KDIM / BK)   // 144

// ---------------- kernel 1: style modulation s[b,ci] ----------------
__global__ __launch_bounds__(512)
void modulate_kernel(const float* __restrict__ style, const float* __restrict__ mod_w,
                     const float* __restrict__ mod_b, float* __restrict__ s) {
    const int b = blockIdx.x, ci = threadIdx.x;
    const float* st = style + b * 512;
    const float* mw = mod_w + ci * 512;
    float acc = 0.f;
#pragma unroll 8
    for (int j = 0; j < 512; ++j) acc += st[j] * mw[j];
    s[b * 512 + ci] = acc * 0.04419417382415922f + mod_b[ci];  // 1/sqrt(512)
}

// ---------------- kernel 2: wsq[co,ci] = sum_k weight^2 ----------------
__global__ __launch_bounds__(512)
void wsq_kernel(const float* __restrict__ weight, float* __restrict__ wsq) {
    const int co = blockIdx.x, ci = threadIdx.x;
    const float* w = weight + (size_t)(co * 512 + ci) * 9;
    float acc = 0.f;
#pragma unroll
    for (int t = 0; t < 9; ++t) acc += w[t] * w[t];
    wsq[co * 512 + ci] = acc;
}

// ---------------- kernel 3: odscale[b,co] = w_scale * demod ----------------
__global__ __launch_bounds__(512)
void demod_kernel(const float* __restrict__ s, const float* __restrict__ wsq,
                  float* __restrict__ odscale) {
    __shared__ float s2[512];
    const int b = blockIdx.x, co = threadIdx.x;
    const float sv = s[b * 512 + co];
    s2[co] = sv * sv;
    __syncthreads();
    const float* wq = wsq + co * 512;
    float acc = 0.f;
#pragma unroll 4
    for (int ci = 0; ci < 512; ++ci) acc += s2[ci] * wq[ci];
    const float w_scale = 0.014731391274719739f;  // 1/sqrt(512*9)
    odscale[b * 512 + co] = w_scale * rsqrtf(w_scale * w_scale * acc + 1e-8f);
}

// ---------------- kernel 4: async-DMA + double-buffered implicit-GEMM conv ----------------
__global__ __launch_bounds__(256)
void conv_wmma_kernel(const float* __restrict__ x, const float* __restrict__ weight,
                      const float* __restrict__ s, const float* __restrict__ odscale,
                      float* __restrict__ out) {
    // As row stride 36 floats = 144 B (16B-aligned rows, distinct banks for v2f frag reads)
    __shared__ __align__(16) float As[2][BM][BK + 4];
    __shared__ __align__(16) float Bs[2][BK][BN + 4];

    const int b  = blockIdx.z;
    const int m0 = blockIdx.y * BM;
    const int n0 = blockIdx.x * BN;

    const int tid  = threadIdx.x;
    const int lane = tid & 31;
    const int wave = tid >> 5;
    const int wm   = (wave >> 2) * 64;    // wave tile 64(M) x 32(N)
    const int wn   = (wave & 3) * 32;
    const int lm   = lane & 15;
    const int lk   = (lane >> 4) * 2;

    // A-staging geometry (thread-invariant): 16B chunk id = tid + 256*i
    const int arow = tid >> 3;            // +32*i
    const int ac4  = (tid & 7) * 4;

    // B-staging geometry (thread-invariant): element id = tid + 256*i -> nn fixed
    const int kkb  = tid >> 7;            // 0/1, +2*i
    const int nn   = tid & 127;
    const int oh   = (n0 + nn) >> 6;
    const int ow   = (n0 + nn) & 63;

    const float* xb = x + (size_t)b * CIN * NN;
    const float* sb = s + b * 512;

    v8f acc[4][2];
#pragma unroll
    for (int i = 0; i < 4; ++i)
#pragma unroll
        for (int j = 0; j < 2; ++j)
#pragma unroll
            for (int e = 0; e < 8; ++e) acc[i][j][e] = 0.f;

    float pb[16];

    // A tile: dense aligned copy -> async DMA straight into LDS (ASYNCcnt)
    auto async_A = [&](int t, int buf) {
        const int k0 = t * BK;
#pragma unroll
        for (int i = 0; i < 4; ++i) {
            const float* g = weight + (size_t)(m0 + arow + 32 * i) * KDIM + k0 + ac4;
            const uint32_t l = (uint32_t)(uintptr_t)&As[buf][arow + 32 * i][ac4];
            asm volatile("global_load_async_to_lds_b128 %0, %1, off"
                         :: "v"(l), "v"(g) : "memory");
        }
    };

    // B tile: im2col gather with zero-pad + s[b,ci] modulation -> registers
    auto load_B = [&](int t) {
        const int k0 = t * BK;
#pragma unroll
        for (int i = 0; i < 16; ++i) {
            const int gk = k0 + kkb + 2 * i;
            const int ci = gk / 9;
            const int r  = gk - ci * 9;
            const int kh = r / 3;
            const int kw = r - kh * 3;
            const int ih = oh + kh - 1;
            const int iw = ow + kw - 1;
            float v = 0.f;
            if ((unsigned)ih < 64u && (unsigned)iw < 64u)
                v = xb[(size_t)ci * NN + ih * 64 + iw] * sb[ci];
            pb[i] = v;
        }
    };

    auto store_B = [&](int buf) {
#pragma unroll
        for (int i = 0; i < 16; ++i)
            Bs[buf][kkb + 2 * i][nn] = pb[i];
    };

    auto compute = [&](int buf) {
#pragma unroll
        for (int ks = 0; ks < 8; ++ks) {
            const int kb = ks * 4 + lk;
            v2f a[4], bf[2];
#pragma unroll
            for (int i = 0; i < 4; ++i)
                a[i] = *(const v2f*)&As[buf][wm + i * 16 + lm][kb];
#pragma unroll
            for (int j = 0; j < 2; ++j) {
                bf[j].x = Bs[buf][kb + 0][wn + j * 16 + lm];
                bf[j].y = Bs[buf][kb + 1][wn + j * 16 + lm];
            }
#pragma unroll
            for (int i = 0; i < 4; ++i)
#pragma unroll
                for (int j = 0; j < 2; ++j)
                    acc[i][j] = __builtin_amdgcn_wmma_f32_16x16x4_f32(
                        false, a[i], false, bf[j], (short)0, acc[i][j], false, false);
        }
    };

    // ---- software pipeline: DMA A(t+1) + reg-prefetch B(t+1) overlap WMMA(t) ----
    async_A(0, 0);
    load_B(0);
    store_B(0);
    asm volatile("s_wait_asynccnt 0x0" ::: "memory");
    __syncthreads();

    for (int t = 0; t < KTILES; ++t) {
        if (t + 1 < KTILES) {
            async_A(t + 1, (t + 1) & 1);
            load_B(t + 1);
        }
        compute(t & 1);
        if (t + 1 < KTILES) store_B((t + 1) & 1);
        asm volatile("s_wait_asynccnt 0x0" ::: "memory");
        __syncthreads();
    }

    // ---- store with demod*w_scale applied per (b, co) ----
#pragma unroll
    for (int i = 0; i < 4; ++i) {
        const int mbase = m0 + wm + i * 16 + ((lane >> 4) << 3);
        float sc[8];
#pragma unroll
        for (int r = 0; r < 8; ++r) sc[r] = odscale[b * 512 + mbase + r];
#pragma unroll
        for (int j = 0; j < 2; ++j) {
            const int n = n0 + wn + j * 16 + lm;
#pragma unroll
            for (int r = 0; r < 8; ++r)
                out[((size_t)b * COUT + mbase + r) * NN + n] = acc[i][j][r] * sc[r];
        }
    }
}

extern "C" void kernel_launch(void* const* d_in, const int* in_sizes, int n_in,
                              void* d_out, int out_size, void* d_ws, size_t ws_size,
                              hipStream_t stream) {
    const float* x      = (const float*)d_in[0];
    const float* style  = (const float*)d_in[1];
    const float* weight = (const float*)d_in[2];
    const float* mod_w  = (const float*)d_in[3];
    const float* mod_b  = (const float*)d_in[4];
    float* out = (float*)d_out;

    float* s       = (float*)d_ws;          // 16*512
    float* odscale = s + 16 * 512;          // 16*512
    float* wsq     = odscale + 16 * 512;    // 512*512

    modulate_kernel<<<16, 512, 0, stream>>>(style, mod_w, mod_b, s);
    wsq_kernel<<<512, 512, 0, stream>>>(weight, wsq);
    demod_kernel<<<16, 512, 0, stream>>>(s, wsq, odscale);

    dim3 grid(NN / BN, COUT / BM, B_);
    conv_wmma_kernel<<<grid, 256, 0, stream>>>(x, weight, s, odscale, out);
}